// CFGCN_89043261981079
// MI455X (gfx1250) — compile-verified
//
#include <hip/hip_runtime.h>
#include <math.h>

typedef __attribute__((ext_vector_type(2))) float v2f;
typedef __attribute__((ext_vector_type(8))) float v8f;

#define N_USERS_C 100000
#define D_C 64
#define LAYERS 3

// ---------------- zero nh accumulator + loss/reg accumulators ----------------
__global__ void k_zero(float* __restrict__ nh, float* __restrict__ acc, long n4) {
  long i = (long)blockIdx.x * blockDim.x + threadIdx.x;
  if (i < 2) acc[i] = 0.0f;
  if (i < n4) ((float4*)nh)[i] = make_float4(0.f, 0.f, 0.f, 0.f);
}

// ---- gather ego embeddings for (users, pos, neg) into buf; accumulate L2 reg ----
__global__ void k_init_buf(const float* __restrict__ emb,
                           const int* __restrict__ users, const int* __restrict__ pos,
                           const int* __restrict__ neg,
                           float* __restrict__ buf, float* __restrict__ acc, int Bb) {
  int t = blockIdx.x * blockDim.x + threadIdx.x;     // 3*B*16 threads, float4 each
  int total = 3 * Bb * 16;
  float ss = 0.0f;
  if (t < total) {
    int j = t >> 4, c = (t & 15) << 2;
    int idx;
    if (j < Bb)            idx = users[j];
    else if (j < 2 * Bb)   idx = pos[j - Bb] + N_USERS_C;
    else                   idx = neg[j - 2 * Bb] + N_USERS_C;
    float4 v = *(const float4*)(emb + (size_t)idx * D_C + c);
    *(float4*)(buf + (size_t)j * D_C + c) = v;
    ss = v.x * v.x + v.y * v.y + v.z * v.z + v.w * v.w;
  }
  for (int off = 16; off; off >>= 1) ss += __shfl_xor(ss, off, 32);
  if ((threadIdx.x & 31) == 0) atomicAdd(acc + 1, ss);
}

// ---- SpMM scatter: nh[dst] += h[src] * sqrt_degree[src]; 16 lanes / edge ----
__global__ void k_scatter(const float* __restrict__ hin, float* __restrict__ nh,
                          const int* __restrict__ src, const int* __restrict__ dst,
                          const float* __restrict__ sd, long total) {
  long t = (long)blockIdx.x * blockDim.x + threadIdx.x;
  if (t >= total) return;
  long e = t >> 4;
  int c = (int)(t & 15) << 2;
  int s = src[e], dvtx = dst[e];
  float w = sd[s];
  float4 v = *(const float4*)(hin + (size_t)s * D_C + c);
  float* o = nh + (size_t)dvtx * D_C + c;
  atomicAdd(o + 0, v.x * w);
  atomicAdd(o + 1, v.y * w);
  atomicAdd(o + 2, v.z * w);
  atomicAdd(o + 3, v.w * w);
}

// ---- h = nh * sqrt_degree[row]; nh = 0 (ready for next layer) ----
__global__ void k_finish(float* __restrict__ h, float* __restrict__ nh,
                         const float* __restrict__ sd, long n4) {
  long i = (long)blockIdx.x * blockDim.x + threadIdx.x;
  if (i >= n4) return;
  long row = i >> 4;                 // 16 float4 per 64-wide row
  float w = sd[row];
  float4 v = ((float4*)nh)[i];
  v.x *= w; v.y *= w; v.z *= w; v.w *= w;
  ((float4*)h)[i] = v;
  ((float4*)nh)[i] = make_float4(0.f, 0.f, 0.f, 0.f);
}

// ---- buf[j] += h[sampled_idx(j)]  (layer-sum, only at sampled rows) ----
__global__ void k_sample_add(const float* __restrict__ h,
                             const int* __restrict__ users, const int* __restrict__ pos,
                             const int* __restrict__ neg,
                             float* __restrict__ buf, int Bb) {
  int t = blockIdx.x * blockDim.x + threadIdx.x;
  int total = 3 * Bb * 16;
  if (t >= total) return;
  int j = t >> 4, c = (t & 15) << 2;
  int idx;
  if (j < Bb)            idx = users[j];
  else if (j < 2 * Bb)   idx = pos[j - Bb] + N_USERS_C;
  else                   idx = neg[j - 2 * Bb] + N_USERS_C;
  float4 v = *(const float4*)(h + (size_t)idx * D_C + c);
  float* o = buf + (size_t)j * D_C + c;
  float4 b = *(float4*)o;
  b.x += v.x; b.y += v.y; b.z += v.z; b.w += v.w;
  *(float4*)o = b;
}

// ---- BPR scores via WMMA: per 16-row tile, diag(U*P^T) and diag(U*N^T) ----
// buf layout: [0,B) users_acc, [B,2B) pos_acc, [2B,3B) neg_acc, each row 64 f32
// final = acc/4  =>  score = dot(u_acc, p_acc) / 16
__global__ void k_score(const float* __restrict__ buf, float* __restrict__ acc, int Bb) {
  int lane = threadIdx.x & 31;
  int wave = (blockIdx.x * blockDim.x + threadIdx.x) >> 5;
  int ntiles = Bb / 16;
  float sp = 0.0f;
  if (wave < ntiles) {                       // uniform per wave; exec stays all-ones
    const float* U  = buf + (size_t)wave * 16 * D_C;
    const float* P  = buf + (size_t)Bb * D_C + (size_t)wave * 16 * D_C;
    const float* Ng = buf + (size_t)2 * Bb * D_C + (size_t)wave * 16 * D_C;
    int row = lane & 15;
    int half2 = (lane >> 4) << 1;            // 0 or 2: K sub-offset for upper half-wave
    v8f dpos = {};
    v8f dneg = {};
#pragma unroll
    for (int k = 0; k < 16; ++k) {
      int col = k * 4 + half2;
      v2f a  = *(const v2f*)(U  + row * D_C + col);   // A frag 16x4 f32
      v2f bp = *(const v2f*)(P  + row * D_C + col);   // B frag 4x16 f32 (= P^T chunk)
      v2f bn = *(const v2f*)(Ng + row * D_C + col);
      dpos = __builtin_amdgcn_wmma_f32_16x16x4_f32(false, a, false, bp,
                                                   (short)0, dpos, false, false);
      dneg = __builtin_amdgcn_wmma_f32_16x16x4_f32(false, a, false, bn,
                                                   (short)0, dneg, false, false);
    }
    // Diagonal: lanes 0..7 hold S[m][m] in d[m]; lanes 24..31 hold S[m][m] in d[m-8]
    float z = 0.0f;
#pragma unroll
    for (int r = 0; r < 8; ++r) {
      bool sel = (lane == r) || (lane == 24 + r);
      z += sel ? (dneg[r] - dpos[r]) : 0.0f;
    }
    bool active = (lane < 8) || (lane >= 24);
    z *= (1.0f / 16.0f);                     // (acc/4)·(acc/4)
    if (active) sp = fmaxf(z, 0.0f) + log1pf(expf(-fabsf(z)));  // stable softplus
  }
  for (int off = 16; off; off >>= 1) sp += __shfl_xor(sp, off, 32);
  if (lane == 0) atomicAdd(acc + 0, sp);
}

__global__ void k_final(const float* __restrict__ acc, float* __restrict__ out, float invB) {
  if (threadIdx.x == 0 && blockIdx.x == 0)
    out[0] = acc[0] * invB + 0.001f * 0.5f * acc[1] * invB;
}

extern "C" void kernel_launch(void* const* d_in, const int* in_sizes, int n_in,
                              void* d_out, int out_size, void* d_ws, size_t ws_size,
                              hipStream_t stream) {
  const float* emb = (const float*)d_in[0];
  const float* sd  = (const float*)d_in[1];
  const int* src   = (const int*)d_in[2];
  const int* dst   = (const int*)d_in[3];
  const int* users = (const int*)d_in[4];
  const int* pos   = (const int*)d_in[5];
  const int* neg   = (const int*)d_in[6];

  const long NE = in_sizes[2];       // 4,800,000 directed edges
  const long Nn = in_sizes[1];       // 150,000 nodes
  const int  Bb = in_sizes[4];       // 8192
  const long ND = Nn * D_C;
  const long n4 = ND >> 2;

  // workspace: h | nh | sampled-buf(3B x 64) | acc[2]
  float* h   = (float*)d_ws;
  float* nh  = h + ND;
  float* buf = nh + ND;
  float* acc = buf + (size_t)3 * Bb * D_C;

  const int T = 256;
  hipLaunchKernelGGL(k_zero, dim3((unsigned)((n4 + T - 1) / T)), dim3(T), 0, stream,
                     nh, acc, n4);

  int tot_s = 3 * Bb * 16;
  hipLaunchKernelGGL(k_init_buf, dim3((tot_s + T - 1) / T), dim3(T), 0, stream,
                     emb, users, pos, neg, buf, acc, Bb);

  long tot_e = NE * 16;
  const float* hin = emb;            // layer 0 reads ego embeddings directly
  for (int l = 0; l < LAYERS; ++l) {
    hipLaunchKernelGGL(k_scatter, dim3((unsigned)((tot_e + T - 1) / T)), dim3(T), 0, stream,
                       hin, nh, src, dst, sd, tot_e);
    hipLaunchKernelGGL(k_finish, dim3((unsigned)((n4 + T - 1) / T)), dim3(T), 0, stream,
                       h, nh, sd, n4);
    hipLaunchKernelGGL(k_sample_add, dim3((tot_s + T - 1) / T), dim3(T), 0, stream,
                       h, users, pos, neg, buf, Bb);
    hin = h;
  }

  int ntiles = Bb / 16;              // 512 waves, one 16-triple tile each
  hipLaunchKernelGGL(k_score, dim3((ntiles * 32 + T - 1) / T), dim3(T), 0, stream,
                     buf, acc, Bb);
  hipLaunchKernelGGL(k_final, dim3(1), dim3(1), 0, stream,
                     acc, (float*)d_out, 1.0f / (float)Bb);
}